// Regressor3d_66271345377457
// MI455X (gfx1250) — compile-verified
//
#include <hip/hip_runtime.h>
#include <hip/hip_bf16.h>

// ---------------------------------------------------------------------------
// SemGCN hand regressor for MI455X (gfx1250), f32 WMMA 16x16x4 path.
// Big broadcast GEMMs ([64,K]x[K,256], K=65536/32768) are HBM-bound (~200MB
// of f32 weights, ~9us at 23.3TB/s) -> keep f32, use V_WMMA_F32_16X16X4_F32,
// split K across waves for occupancy, deterministic chunk reduction.
// ---------------------------------------------------------------------------

typedef float v2f __attribute__((ext_vector_type(2)));
typedef float v8f __attribute__((ext_vector_type(8)));

#define NNODES 21
#define BATCH 64
#define ROWS (BATCH * NNODES)   // 1344 = 84 * 16

// Hand-skeleton adjacency (eye + edges), row bitmasks (bit m set => A[n][m] allowed)
__device__ __constant__ unsigned int ADJ_ROW[NNODES] = {
    139811u,  7u,     14u,     28u,     24u,
    97u,      224u,   448u,    384u,    1537u,
    3584u,    7168u,  6144u,   24577u,  57344u,
    114688u,  98304u, 393217u, 917504u, 1835008u,
    1572864u
};

__device__ __forceinline__ v8f wmma_f32_4(v2f a, v2f b, v8f c) {
    // D = A(16x4 f32) * B(4x16 f32) + C(16x16 f32)
    return __builtin_amdgcn_wmma_f32_16x16x4_f32(
        /*neg_a=*/false, a, /*neg_b=*/false, b,
        /*c_mod=*/(short)0, c, /*reuse_a=*/false, /*reuse_b=*/false);
}

// ---------------------------------------------------------------------------
// Big broadcast GEMM: P[wsel][chunk][64][256] (+=) X[64,K] * W[wsel][K,256]
// over this block's K-chunk. One wave per (ntile, chunk, wsel); the wave owns
// all 4 M-tiles (64 rows) so weights are read exactly once per chunk.
// ---------------------------------------------------------------------------
__global__ void __launch_bounds__(32)
big_gemm_f32(const float* __restrict__ X, const float* __restrict__ W,
             float* __restrict__ P, int K, int chunkK) {
    const int ntile = blockIdx.x;          // 0..15  (N = 256 / 16)
    const int chunk = blockIdx.y;
    const int wsel  = blockIdx.z;          // 0: W[0] (self), 1: W[1] (neigh)
    W += (size_t)wsel * K * 256;
    P += ((size_t)wsel * gridDim.y + chunk) * (64 * 256);

    const int lane  = threadIdx.x;
    const int l15   = lane & 15;
    const int khalf = lane >> 4;           // 0 or 1
    const int ncol  = ntile * 16 + l15;

    v8f acc0 = {}, acc1 = {}, acc2 = {}, acc3 = {};
    const int k0 = chunk * chunkK;
    const int k1 = k0 + chunkK;
    #pragma unroll 2
    for (int k = k0; k < k1; k += 4) {
        const int ka = k + khalf * 2;      // this lane-half's K pair
        v2f b;
        b.x = W[(size_t)ka * 256 + ncol];
        b.y = W[(size_t)(ka + 1) * 256 + ncol];
        v2f a0 = *(const v2f*)&X[(size_t)(l15 +  0) * K + ka];
        v2f a1 = *(const v2f*)&X[(size_t)(l15 + 16) * K + ka];
        v2f a2 = *(const v2f*)&X[(size_t)(l15 + 32) * K + ka];
        v2f a3 = *(const v2f*)&X[(size_t)(l15 + 48) * K + ka];
        acc0 = wmma_f32_4(a0, b, acc0);
        acc1 = wmma_f32_4(a1, b, acc1);
        acc2 = wmma_f32_4(a2, b, acc2);
        acc3 = wmma_f32_4(a3, b, acc3);
    }
    #pragma unroll
    for (int r = 0; r < 8; ++r) {
        const int m = r + khalf * 8;       // C/D layout: lanes16-31 hold M=r+8
        P[(size_t)(m +  0) * 256 + ncol] = acc0[r];
        P[(size_t)(m + 16) * 256 + ncol] = acc1[r];
        P[(size_t)(m + 32) * 256 + ncol] = acc2[r];
        P[(size_t)(m + 48) * 256 + ncol] = acc3[r];
    }
}

// Deterministic reduction of per-chunk partials: H[i] = sum_c P[c][i]
__global__ void reduce_chunks(const float* __restrict__ P, float* __restrict__ H,
                              int chunks) {
    const int idx = blockIdx.x * 256 + threadIdx.x;    // < 64*256
    float s = 0.f;
    for (int c = 0; c < chunks; ++c) s += P[(size_t)c * (64 * 256) + idx];
    H[idx] = s;
}

// ---------------------------------------------------------------------------
// Small GEMM: H[wsel][1344,Cout] = X[1344, Cin (row stride ldx)] * W[wsel][Cin,Cout]
// One wave per 16x16 output tile. N guarded for Cout not multiple of 16 (Cout=3).
// ---------------------------------------------------------------------------
__global__ void __launch_bounds__(32)
small_gemm_f32(const float* __restrict__ X, int ldx,
               const float* __restrict__ W, int Cin, int Cout,
               float* __restrict__ H) {
    const int mtile = blockIdx.x;          // 0..83
    const int ntile = blockIdx.y;
    const int wsel  = blockIdx.z;
    W += (size_t)wsel * Cin * Cout;
    H += (size_t)wsel * ROWS * Cout;

    const int lane  = threadIdx.x;
    const int l15   = lane & 15;
    const int khalf = lane >> 4;
    const int mrow  = mtile * 16 + l15;
    const int n     = ntile * 16 + l15;
    const bool nok  = (n < Cout);

    v8f acc = {};
    #pragma unroll 4
    for (int k = 0; k < Cin; k += 4) {
        const int ka = k + khalf * 2;
        v2f a = *(const v2f*)&X[(size_t)mrow * ldx + ka];
        v2f b;
        b.x = nok ? W[(size_t)ka * Cout + n] : 0.f;
        b.y = nok ? W[(size_t)(ka + 1) * Cout + n] : 0.f;
        acc = wmma_f32_4(a, b, acc);
    }
    #pragma unroll
    for (int r = 0; r < 8; ++r) {
        const int m = r + khalf * 8;
        if (nok) H[(size_t)(mtile * 16 + m) * Cout + n] = acc[r];
    }
}

// ---------------------------------------------------------------------------
// Masked row-softmax of learnable adjacency logits e[21,21] -> A[21,21].
// Matches jax.nn.softmax(where(mask, e, -9e15), axis=1): unmasked -> 0.
// ---------------------------------------------------------------------------
__global__ void adj_softmax(const float* __restrict__ e, float* __restrict__ A) {
    const int n = threadIdx.x;
    if (n >= NNODES) return;
    const unsigned rm = ADJ_ROW[n];
    float mx = -3.4e38f;
    for (int m = 0; m < NNODES; ++m)
        if ((rm >> m) & 1u) mx = fmaxf(mx, e[n * NNODES + m]);
    float vals[NNODES];
    float s = 0.f;
    for (int m = 0; m < NNODES; ++m) {
        float v = ((rm >> m) & 1u) ? __expf(e[n * NNODES + m] - mx) : 0.f;
        vals[m] = v; s += v;
    }
    const float inv = 1.f / s;
    for (int m = 0; m < NNODES; ++m) A[n * NNODES + m] = vals[m] * inv;
}

// Broadcast-input layers: h0/h1 identical across nodes, softmax rows sum to 1:
// y[b,n,d] = A[n,n]*H0[b,d] + (1-A[n,n])*H1[b,d] + bias[d]
__global__ void uniform_mix(const float* __restrict__ H0, const float* __restrict__ H1,
                            const float* __restrict__ bias, const float* __restrict__ A,
                            float* __restrict__ Y) {
    const int idx = blockIdx.x * 256 + threadIdx.x;    // < 64*21*256
    const int d = idx & 255;
    const int n = (idx >> 8) % NNODES;
    const int b = idx / (NNODES * 256);
    const float ad = A[n * NNODES + n];
    Y[idx] = ad * H0[b * 256 + d] + (1.f - ad) * H1[b * 256 + d] + bias[d];
}

// General mixing: y[b,n,d] = A[n,n]*h0[b,n,d] + sum_{m!=n} A[n,m]*h1[b,m,d] + bias[d]
// (A[n,m]==0 for non-neighbors, so the full sum minus the diagonal is exact.)
__global__ void mix_general(const float* __restrict__ H0, const float* __restrict__ H1,
                            const float* __restrict__ bias, const float* __restrict__ A,
                            float* __restrict__ Y, int C) {
    const int idx = blockIdx.x * 256 + threadIdx.x;
    if (idx >= ROWS * C) return;
    const int d   = idx % C;
    const int row = idx / C;         // b*21 + n
    const int n   = row % NNODES;
    const int b   = row / NNODES;
    float acc = A[n * NNODES + n] * H0[(size_t)row * C + d];
    const float* h1b = H1 + (size_t)b * NNODES * C + d;
    #pragma unroll
    for (int m = 0; m < NNODES; ++m) {
        if (m == n) continue;
        acc += A[n * NNODES + m] * h1b[(size_t)m * C];
    }
    Y[idx] = acc + bias[d];
}

// ---------------------------------------------------------------------------
// BatchNorm (training stats over B*N=1344 samples) + ReLU (+ optional residual),
// one block per channel: deterministic LDS tree reduction.
// out[i*ld_out + d] = (resid ? resid[i*ld_res + d] : 0) + relu(bn(Y[i*C+d]))
// ---------------------------------------------------------------------------
__global__ void __launch_bounds__(256)
bn_relu(const float* __restrict__ Y, const float* __restrict__ gamma,
        const float* __restrict__ beta, const float* __restrict__ resid, int ld_res,
        float* __restrict__ out, int ld_out, int C) {
    const int d = blockIdx.x;
    __shared__ float s1[256], s2[256];
    float sum = 0.f, sumsq = 0.f;
    for (int i = threadIdx.x; i < ROWS; i += 256) {
        const float v = Y[(size_t)i * C + d];
        sum += v; sumsq += v * v;
    }
    s1[threadIdx.x] = sum; s2[threadIdx.x] = sumsq;
    __syncthreads();
    for (int s = 128; s > 0; s >>= 1) {
        if (threadIdx.x < s) {
            s1[threadIdx.x] += s1[threadIdx.x + s];
            s2[threadIdx.x] += s2[threadIdx.x + s];
        }
        __syncthreads();
    }
    const float mu    = s1[0] * (1.f / ROWS);
    const float var   = s2[0] * (1.f / ROWS) - mu * mu;
    const float scale = rsqrtf(var + 1e-5f) * gamma[d];
    const float shift = beta[d] - mu * scale;
    for (int i = threadIdx.x; i < ROWS; i += 256) {
        float v = fmaxf(Y[(size_t)i * C + d] * scale + shift, 0.f);
        if (resid) v += resid[(size_t)i * ld_res + d];
        out[(size_t)i * ld_out + d] = v;
    }
}

// ---------------------------------------------------------------------------
// Host orchestration
// ---------------------------------------------------------------------------
// Workspace layout (float offsets)
static constexpr size_t OFF_P3   = 0;                        // 2*32*16384
static constexpr size_t OFF_P4   = OFF_P3 + 2u*32*16384;     // 2*16*16384
static constexpr size_t OFF_H3   = OFF_P4 + 2u*16*16384;     // 2*16384
static constexpr size_t OFF_H4   = OFF_H3 + 2u*16384;        // 2*16384
static constexpr size_t OFF_A    = OFF_H4 + 2u*16384;        // 512
static constexpr size_t OFF_X3   = OFF_A  + 512;             // 1344*256
static constexpr size_t OFF_X4   = OFF_X3 + (size_t)ROWS*256;
static constexpr size_t OFF_FEAT = OFF_X4 + (size_t)ROWS*256; // 1344*512
static constexpr size_t OFF_T    = OFF_FEAT + (size_t)ROWS*512;
static constexpr size_t OFF_H0   = OFF_T  + (size_t)ROWS*256; // h0|h1 contiguous
static constexpr size_t OFF_Y    = OFF_H0 + 2u*(size_t)ROWS*256;
static constexpr size_t OFF_FA   = OFF_Y  + (size_t)ROWS*256;
static constexpr size_t OFF_FB   = OFF_FA + (size_t)ROWS*256;
// total ~5.08M floats (~20.3 MB) — within typical workspace budgets.

extern "C" void kernel_launch(void* const* d_in, const int* in_sizes, int n_in,
                              void* d_out, int out_size, void* d_ws, size_t ws_size,
                              hipStream_t stream) {
    (void)in_sizes; (void)n_in; (void)out_size; (void)ws_size;
    const float* out4 = (const float*)d_in[3];   // [64, 65536]
    const float* out5 = (const float*)d_in[4];   // [64, 32768]
    // params flattened: layer i at d_in[5 + 5*i + {W,e,b,gamma,beta}]
    // order: gconv3,res3a,res3b,gconv4,res4a,res4b,gconv5,res5a,res5b,res6a,res6b,gconvout
    auto P  = [&](int layer, int field) { return (const float*)d_in[5 + 5 * layer + field]; };

    float* ws = (float*)d_ws;
    float* A    = ws + OFF_A;
    float* H3   = ws + OFF_H3;   // [2][64,256]
    float* H4   = ws + OFF_H4;
    float* X3   = ws + OFF_X3;
    float* X4   = ws + OFF_X4;
    float* FEAT = ws + OFF_FEAT;
    float* T    = ws + OFF_T;
    float* H0   = ws + OFF_H0;   // [2][1344,Cout]
    float* Y    = ws + OFF_Y;
    float* FA   = ws + OFF_FA;
    float* FB   = ws + OFF_FB;

    // ---- Stage 1: big broadcast GEMMs (K split for occupancy) ----
    const int CK = 2048;
    big_gemm_f32<<<dim3(16, 65536 / CK, 2), 32, 0, stream>>>(out4, P(0, 0), ws + OFF_P3, 65536, CK);
    big_gemm_f32<<<dim3(16, 32768 / CK, 2), 32, 0, stream>>>(out5, P(3, 0), ws + OFF_P4, 32768, CK);
    reduce_chunks<<<64, 256, 0, stream>>>(ws + OFF_P3,                         H3,         32);
    reduce_chunks<<<64, 256, 0, stream>>>(ws + OFF_P3 + 32u * 16384,           H3 + 16384, 32);
    reduce_chunks<<<64, 256, 0, stream>>>(ws + OFF_P4,                         H4,         16);
    reduce_chunks<<<64, 256, 0, stream>>>(ws + OFF_P4 + 16u * 16384,           H4 + 16384, 16);

    // broadcast-layer epilogue: adjacency softmax + uniform mix + BN/ReLU
    auto run_uniform = [&](const float* Hh, int layer, float* out) {
        adj_softmax<<<1, 32, 0, stream>>>(P(layer, 1), A);
        uniform_mix<<<ROWS, 256, 0, stream>>>(Hh, Hh + 16384, P(layer, 2), A, Y);
        bn_relu<<<256, 256, 0, stream>>>(Y, P(layer, 3), P(layer, 4),
                                         nullptr, 0, out, 256, 256);
    };
    // full sem_gconv layer: softmax + 2x small GEMM + mix + BN/ReLU (+residual)
    auto run_layer = [&](const float* X, int ldx, int layer, int Cin, int Cout,
                         const float* resid, int ld_res, float* out, int ld_out) {
        adj_softmax<<<1, 32, 0, stream>>>(P(layer, 1), A);
        small_gemm_f32<<<dim3(ROWS / 16, (Cout + 15) / 16, 2), 32, 0, stream>>>(
            X, ldx, P(layer, 0), Cin, Cout, H0);
        const int total = ROWS * Cout;
        mix_general<<<(total + 255) / 256, 256, 0, stream>>>(
            H0, H0 + (size_t)ROWS * Cout, P(layer, 2), A, Y, Cout);
        bn_relu<<<Cout, 256, 0, stream>>>(Y, P(layer, 3), P(layer, 4),
                                          resid, ld_res, out, ld_out, Cout);
    };

    // ---- x3 branch: gconv3 -> res3a -> res3b (+x3) -> FEAT[:, 0:256] ----
    run_uniform(H3, /*gconv3*/ 0, X3);
    run_layer(X3, 256, /*res3a*/ 1, 256, 128, nullptr, 0, T, 128);
    run_layer(T, 128, /*res3b*/ 2, 128, 256, X3, 256, FEAT, 512);

    // ---- x4 branch: gconv4 -> res4a -> res4b (+x4) -> FEAT[:, 256:512] ----
    run_uniform(H4, /*gconv4*/ 3, X4);
    run_layer(X4, 256, /*res4a*/ 4, 256, 128, nullptr, 0, T, 128);
    run_layer(T, 128, /*res4b*/ 5, 128, 256, X4, 256, FEAT + 256, 512);

    // ---- trunk: gconv5 -> res5 -> res6 -> gconvout ----
    run_layer(FEAT, 512, /*gconv5*/ 6, 512, 256, nullptr, 0, FA, 256);
    run_layer(FA, 256, /*res5a*/ 7, 256, 128, nullptr, 0, T, 128);
    run_layer(T, 128, /*res5b*/ 8, 128, 256, FA, 256, FB, 256);
    run_layer(FB, 256, /*res6a*/ 9, 256, 128, nullptr, 0, T, 128);
    run_layer(T, 128, /*res6b*/ 10, 128, 256, FB, 256, FA, 256);
    run_layer(FA, 256, /*gconvout*/ 11, 256, 3, nullptr, 0, (float*)d_out, 3);
}